// MultistablePhysicallyImplementableRandomizedOscillatorsNetwork_80582176407996
// MI455X (gfx1250) — compile-verified
//
#include <hip/hip_runtime.h>
#include <hip/hip_bf16.h>
#include <stdint.h>

// MPIRON oscillator recurrence for MI455X (gfx1250), wave32, f32 WMMA 16x16x4.
//
// Persistent kernel: 128 WGs x 256 threads (8 waves). WG (msuper, n) owns state
// rows [32*msuper, 32*msuper+32) x cols [16n, 16n+16)  (two 16x16 WMMA tiles).
// Waves 0-3 / 4-7 split K=1024 for tile 0 / 1; LDS tree-reduce per tile.
//
// Weight strips (h2h[:, 16n:16n+16], h2hinv[...], x2h[...]) are invariant over
// all 512 steps -> staged ONCE into LDS (row-major [K][16], conflict-free for
// the 16x16x4 B-fragment pattern) via GLOBAL_LOAD_ASYNC_TO_LDS_B128 +
// s_wait_asynccnt. Inner loop: 1x global b64 (A from L2-resident state) +
// 1x ds_load_2addr_b32 (B) + 1x v_wmma_f32_16x16x4_f32, with compile-time
// trip counts so the loop control stays scalar (no exec-mask juggling).
//
// Device-wide sync: one-shot atomic barrier counters (2 per step), zeroed by
// hipMemsetAsync each launch => deterministic under graph replay.

typedef float v2f __attribute__((ext_vector_type(2)));
typedef float v8f __attribute__((ext_vector_type(8)));

#define BB   64
#define TT   512
#define HH   1024
#define NIN  128
#define NWG  128
#define DT_C 0.042f

typedef __attribute__((address_space(3))) const float lds_cfloat;

__device__ __forceinline__ unsigned lds_off(const float* p)
{
    return (unsigned)(uintptr_t)(lds_cfloat*)p;
}

// async global->LDS copy of one 16-byte chunk per lane (GLOBAL_LOAD_ASYNC_TO_LDS_B128)
__device__ __forceinline__ void async_copy_b128(unsigned lds_byte_off, const float* gsrc)
{
    asm volatile("global_load_async_to_lds_b128 %0, %1, off"
                 :: "v"(lds_byte_off), "v"(gsrc) : "memory");
}

// KLEN-deep 16x16 WMMA span. Arow/ldsB are already offset to this wave's
// K-chunk; KLEN is compile-time so loop control is scalar and addresses fold
// into immediate offsets.
template <int KLEN, int UNROLL>
__device__ __forceinline__ v8f wmma_span_lds(const float* __restrict__ Arow,
                                             const float* __restrict__ ldsB,
                                             int khalf, int lidx, v8f acc)
{
    #pragma unroll UNROLL
    for (int k = 0; k < KLEN; k += 4) {
        const int kb = k + 2 * khalf;
        v2f a = *(const v2f*)(Arow + kb);
        v2f b;
        b.x = ldsB[kb * 16 + lidx];
        b.y = ldsB[(kb + 1) * 16 + lidx];
        acc = __builtin_amdgcn_wmma_f32_16x16x4_f32(false, a, false, b, (short)0, acc,
                                                    false, false);
    }
    return acc;
}

__device__ __forceinline__ void grid_barrier(unsigned* __restrict__ bar, int slot)
{
    __threadfence();
    __syncthreads();
    if (threadIdx.x == 0) {
        __hip_atomic_fetch_add(&bar[slot], 1u, __ATOMIC_ACQ_REL, __HIP_MEMORY_SCOPE_AGENT);
        unsigned v;
        do {
            __builtin_amdgcn_s_sleep(2);
            v = __hip_atomic_load(&bar[slot], __ATOMIC_ACQUIRE, __HIP_MEMORY_SCOPE_AGENT);
        } while (v < (unsigned)NWG);
    }
    __syncthreads();
    __threadfence();
}

__global__ __launch_bounds__(256, 1)
void mpiron_persistent(const float* __restrict__ x,       // [B,T,NIN]
                       const float* __restrict__ x2h,     // [NIN,H]
                       const float* __restrict__ h2h,     // [H,H]
                       const float* __restrict__ h2hinv,  // [H,H]
                       const float* __restrict__ bias,    // [H]
                       const float* __restrict__ gamma_v, // [H]
                       const float* __restrict__ eps_v,   // [H]
                       float* __restrict__ out,           // [B,T,H] ++ [B,H]
                       float* __restrict__ hybuf,         // [B,H] (zeroed)
                       float* __restrict__ sbuf,          // [B,H]
                       unsigned* __restrict__ bar)        // [2*T] (zeroed)
{
    __shared__ float wH [HH * 16];   // h2h   column strip, row-major [K][16]  (64KB)
    __shared__ float wHI[HH * 16];   // h2hinv column strip                    (64KB)
    __shared__ float wX [NIN * 16];  // x2h   column strip                     (8KB)
    __shared__ float redA[8][8][32]; // per-wave WMMA partials                 (8KB)
    __shared__ float redB[8][8][32]; //                                        (8KB)

    const int tid   = threadIdx.x;
    const int lane  = tid & 31;
    const int wid   = tid >> 5;        // 0..7
    const int khalf = lane >> 4;       // A/B fragment K-half
    const int lidx  = lane & 15;       // M (A) / N (B,C,D)

    const int wg      = blockIdx.x;    // 0..127
    const int msuper  = wg & 1;        // 32-row super block
    const int n       = wg >> 1;       // 16-col block
    const int cn      = n * 16;
    const int tilesel = wid >> 2;      // which 16-row tile this wave works on
    const int kslot   = wid & 3;       // K-chunk within the tile
    const int bm      = msuper * 32 + tilesel * 16;
    const int col     = cn + lidx;
    const int rbase   = tilesel * 4;   // base wave slot for reduction
    const bool isred  = (kslot == 0);  // waves 0 and 4 do the elementwise work

    // ---- one-time async staging of the weight strips into LDS -------------
    {
        const unsigned oH  = lds_off(wH);
        const unsigned oHI = lds_off(wHI);
        const unsigned oX  = lds_off(wX);
        for (int c = tid; c < HH * 4; c += 256) {      // 4096 x 16B chunks
            const int k  = c >> 2;
            const int c4 = (c & 3) << 2;
            async_copy_b128(oH  + (unsigned)c * 16, h2h    + (size_t)k * HH + cn + c4);
            async_copy_b128(oHI + (unsigned)c * 16, h2hinv + (size_t)k * HH + cn + c4);
        }
        for (int c = tid; c < NIN * 4; c += 256) {     // 512 x 16B chunks
            const int k  = c >> 2;
            const int c4 = (c & 3) << 2;
            async_copy_b128(oX + (unsigned)c * 16, x2h + (size_t)k * HH + cn + c4);
        }
        asm volatile("s_wait_asynccnt 0x0" ::: "memory");
    }
    __syncthreads();

    const float bia = bias[col];
    const float gam = gamma_v[col];
    const float epc = eps_v[col];

    float hy[8], hz[8];
    #pragma unroll
    for (int r = 0; r < 8; ++r) { hy[r] = 0.0f; hz[r] = 0.0f; }

    const int kw0 = kslot * (HH / 4);   // 256-wide K chunk (hidden GEMMs)
    const int ki0 = kslot * (NIN / 4);  // 32-wide K chunk (input GEMM)

    // Pre-offset bases: K-loop trip counts become compile-time constants.
    const float* hyA  = hybuf + (size_t)(bm + lidx) * HH + kw0;
    const float* sA   = sbuf  + (size_t)(bm + lidx) * HH + kw0;
    const float* wHk  = wH  + kw0 * 16;
    const float* wHIk = wHI + kw0 * 16;
    const float* wXk  = wX  + ki0 * 16;

    for (int t = 0; t < TT; ++t) {
        // ---------------- phase 1: S = tanh(HY @ h2h + bias) ----------------
        {
            v8f acc = {0.f, 0.f, 0.f, 0.f, 0.f, 0.f, 0.f, 0.f};
            acc = wmma_span_lds<HH / 4, 16>(hyA, wHk, khalf, lidx, acc);
            #pragma unroll
            for (int r = 0; r < 8; ++r) redA[wid][r][lane] = acc[r];
        }
        __syncthreads();
        if (isred) {
            #pragma unroll
            for (int r = 0; r < 8; ++r) {
                float s = redA[rbase + 0][r][lane] + redA[rbase + 1][r][lane] +
                          redA[rbase + 2][r][lane] + redA[rbase + 3][r][lane];
                s = tanhf(s + bia);
                const int row = bm + r + 8 * khalf;
                sbuf[(size_t)row * HH + col] = s;
            }
        }
        grid_barrier(bar, 2 * t);

        // ---------------- phase 2: W = S @ h2hinv ; I = x_t @ x2h ----------
        {
            const float* xA = x + ((size_t)(bm + lidx) * TT + (size_t)t) * NIN + ki0;
            __builtin_prefetch(xA + NIN, 0, 1);   // next step's input row
            v8f accI = {0.f, 0.f, 0.f, 0.f, 0.f, 0.f, 0.f, 0.f};
            accI = wmma_span_lds<NIN / 4, 8>(xA, wXk, khalf, lidx, accI);
            v8f accW = {0.f, 0.f, 0.f, 0.f, 0.f, 0.f, 0.f, 0.f};
            accW = wmma_span_lds<HH / 4, 16>(sA, wHIk, khalf, lidx, accW);
            #pragma unroll
            for (int r = 0; r < 8; ++r) { redA[wid][r][lane] = accW[r];
                                          redB[wid][r][lane] = accI[r]; }
        }
        __syncthreads();
        if (isred) {
            #pragma unroll
            for (int r = 0; r < 8; ++r) {
                const float w  = redA[rbase + 0][r][lane] + redA[rbase + 1][r][lane] +
                                 redA[rbase + 2][r][lane] + redA[rbase + 3][r][lane];
                const float ii = tanhf(redB[rbase + 0][r][lane] + redB[rbase + 1][r][lane] +
                                       redB[rbase + 2][r][lane] + redB[rbase + 3][r][lane]);
                hz[r] += DT_C * (ii + w - gam * hy[r] - epc * hz[r]);
                hy[r] += DT_C * hz[r];
                const int row = bm + r + 8 * khalf;
                out[((size_t)row * TT + (size_t)t) * HH + col] = hy[r];
                hybuf[(size_t)row * HH + col] = hy[r];
                if (t == TT - 1)
                    out[(size_t)BB * TT * HH + (size_t)row * HH + col] = hy[r];
            }
        }
        grid_barrier(bar, 2 * t + 1);
    }
}

extern "C" void kernel_launch(void* const* d_in, const int* in_sizes, int n_in,
                              void* d_out, int out_size, void* d_ws, size_t ws_size,
                              hipStream_t stream)
{
    (void)in_sizes; (void)n_in; (void)out_size; (void)ws_size;

    const float* x       = (const float*)d_in[0]; // [B,T,128]
    const float* x2h     = (const float*)d_in[1]; // [128,1024]
    const float* h2h     = (const float*)d_in[2]; // [1024,1024]
    const float* h2hinv  = (const float*)d_in[3]; // [1024,1024]
    const float* bias    = (const float*)d_in[4]; // [1024]
    const float* gamma_v = (const float*)d_in[5]; // [1024]
    const float* eps_v   = (const float*)d_in[6]; // [1024]

    float*    hybuf = (float*)d_ws;               // 64*1024 f32
    float*    sbuf  = hybuf + (size_t)BB * HH;    // 64*1024 f32
    unsigned* bar   = (unsigned*)(sbuf + (size_t)BB * HH); // 2*T u32

    const size_t clear_bytes = (size_t)BB * HH * sizeof(float) * 2
                             + (size_t)(2 * TT) * sizeof(unsigned);
    hipMemsetAsync(d_ws, 0, clear_bytes, stream);

    mpiron_persistent<<<NWG, 256, 0, stream>>>(x, x2h, h2h, h2hinv,
                                               bias, gamma_v, eps_v,
                                               (float*)d_out, hybuf, sbuf, bar);
}